// svPN_65438121722153
// MI455X (gfx1250) — compile-verified
//
#include <hip/hip_runtime.h>
#include <hip/hip_bf16.h>
#include <math.h>

// Problem constants (match reference)
#define BATCH 256
#define DDIM  256
#define NDIM  1024

typedef float v2f __attribute__((ext_vector_type(2)));
typedef float v8f __attribute__((ext_vector_type(8)));

// ---------------------------------------------------------------------------
// Phase 1: per-batch Gram matrix  G = X * X^T   (X is DDIM x NDIM, row-major,
// n contiguous).  G[i][j] = sum_n x[b][i][n] * x[b][j][n].
// Full-precision fp32 WMMA: V_WMMA_F32_16X16X4_F32, D = A(16x4) x B(4x16) + C.
// Per ISA 7.12.2 fp32 layouts, both A and B operands for our tile are a
// float2 load at  x[row_base + (lane&15)][n + 2*(lane>>4)].
//
// 2x2 register tiling: each wave owns a 32x32 region of G (4 accumulators),
// so each loaded A/B operand feeds two WMMAs -> 1 load per WMMA (vs 2 in the
// naive version).  8x8 = 64 macro-tiles per batch, 8 waves -> 8 macro-tiles
// per wave, fully uniform (EXEC all-ones as WMMA requires).
// ---------------------------------------------------------------------------
__global__ __launch_bounds__(256) void svpn_syrk_wmma(const float* __restrict__ x,
                                                      float* __restrict__ G)
{
    const int b    = blockIdx.x;
    const int tid  = threadIdx.x;
    const int wave = tid >> 5;     // 0..7
    const int lane = tid & 31;
    const int lo   = lane & 15;
    const int hi   = lane >> 4;    // 0 or 1

    const float* xb = x + (size_t)b * (DDIM * (size_t)NDIM);
    float*       Gb = G + (size_t)b * (DDIM * (size_t)DDIM);

    for (int mt = wave; mt < 64; mt += 8) {
        const int i0 = (mt >> 3) << 5;     // 32-row macro tile origin
        const int j0 = (mt & 7) << 5;      // 32-col macro tile origin
        const float* pA0 = xb + (size_t)(i0 + lo) * NDIM + (hi << 1);
        const float* pA1 = pA0 + (size_t)16 * NDIM;
        const float* pB0 = xb + (size_t)(j0 + lo) * NDIM + (hi << 1);
        const float* pB1 = pB0 + (size_t)16 * NDIM;

        v8f acc00 = {0.f, 0.f, 0.f, 0.f, 0.f, 0.f, 0.f, 0.f};
        v8f acc01 = acc00, acc10 = acc00, acc11 = acc00;

        #pragma unroll 2
        for (int n = 0; n < NDIM; n += 4) {
            v2f a0 = *reinterpret_cast<const v2f*>(pA0 + n);
            v2f a1 = *reinterpret_cast<const v2f*>(pA1 + n);
            v2f b0 = *reinterpret_cast<const v2f*>(pB0 + n);
            v2f b1 = *reinterpret_cast<const v2f*>(pB1 + n);
            // 8 args: (neg_a, A, neg_b, B, c_mod, C, reuse_a, reuse_b)
            acc00 = __builtin_amdgcn_wmma_f32_16x16x4_f32(
                false, a0, false, b0, (short)0, acc00, false, false);
            acc01 = __builtin_amdgcn_wmma_f32_16x16x4_f32(
                false, a0, false, b1, (short)0, acc01, false, false);
            acc10 = __builtin_amdgcn_wmma_f32_16x16x4_f32(
                false, a1, false, b0, (short)0, acc10, false, false);
            acc11 = __builtin_amdgcn_wmma_f32_16x16x4_f32(
                false, a1, false, b1, (short)0, acc11, false, false);
        }
        // D layout: VGPR r holds (M = r + 8*hi, N = lo)
        #pragma unroll
        for (int r = 0; r < 8; ++r) {
            const int m = r + (hi << 3);
            Gb[(size_t)(i0 + m)      * DDIM + (j0 + lo)]      = acc00[r];
            Gb[(size_t)(i0 + m)      * DDIM + (j0 + 16 + lo)] = acc01[r];
            Gb[(size_t)(i0 + 16 + m) * DDIM + (j0 + lo)]      = acc10[r];
            Gb[(size_t)(i0 + 16 + m) * DDIM + (j0 + 16 + lo)] = acc11[r];
        }
    }
}

// ---------------------------------------------------------------------------
// Phase 2: power iteration entirely in G-space (tiny). One block per batch.
//   v_new = norm(G v);  deflation: G1 = G - s1^2 v1f v1f^T
//   s  = ||w||^2 / max(||w||,eps)  with  w = G v_pre / nu,  nu = max(sqrt(v^T G v), eps)
// Emits per batch: v1p, v1f, v2p, v2f (vecs) and s1, nu1, s2, nu2, c12 (scal).
// Matvec uses G's symmetry for coalescing: row-dot == column read G[j*D + t].
// ---------------------------------------------------------------------------
__device__ __forceinline__ float block_reduce_sum(float val, float* red, int t)
{
    __syncthreads();           // protect prior use of red / prior v reads
    red[t] = val;
    __syncthreads();
    for (int s = 128; s > 0; s >>= 1) {
        if (t < s) red[t] += red[t + s];
        __syncthreads();
    }
    return red[0];
}

__global__ __launch_bounds__(256) void svpn_power(const float* __restrict__ G,
                                                  const float* __restrict__ weight,
                                                  float* __restrict__ vecs,   // [B][4][DDIM]
                                                  float* __restrict__ scal)   // [B][8]
{
    __shared__ float v[DDIM];
    __shared__ float red[DDIM];
    const int b = blockIdx.x;
    const int t = threadIdx.x;
    const float* Gb = G + (size_t)b * DDIM * DDIM;
    const float  EPSN = 1e-12f;

    // ---------------- singular component 1 ----------------
    v[t] = weight[t];
    __syncthreads();
    for (int it = 0; it < 2; ++it) {           // v1p = norm(G norm(G w0))
        float w = 0.f;
        for (int j = 0; j < DDIM; ++j) w = fmaf(Gb[(size_t)j * DDIM + t], v[j], w);
        float nr = fmaxf(sqrtf(block_reduce_sum(w * w, red, t)), EPSN);
        v[t] = w / nr;
        __syncthreads();
    }
    vecs[((size_t)b * 4 + 0) * DDIM + t] = v[t];              // v1p

    float a1 = 0.f;                                           // a1 = G v1p
    for (int j = 0; j < DDIM; ++j) a1 = fmaf(Gb[(size_t)j * DDIM + t], v[j], a1);
    float nu1sq = block_reduce_sum(v[t] * a1, red, t);        // v1p^T G v1p
    float nu1   = fmaxf(sqrtf(fmaxf(nu1sq, 0.f)), EPSN);
    float w1    = a1 / nu1;                                   // w1 = X^T u1
    float wn1   = sqrtf(block_reduce_sum(w1 * w1, red, t));
    float s1    = wn1 * wn1 / fmaxf(wn1, EPSN);               // s1 = u1^T X v1f
    float v1f   = w1 / fmaxf(wn1, EPSN);
    vecs[((size_t)b * 4 + 1) * DDIM + t] = v1f;

    // ---------------- singular component 2 (G1 = G - s1^2 v1f v1f^T) -------
    const float s1sq = s1 * s1;
    __syncthreads();
    v[t] = weight[t];
    __syncthreads();
    for (int it = 0; it < 2; ++it) {
        float w = 0.f;
        for (int j = 0; j < DDIM; ++j) w = fmaf(Gb[(size_t)j * DDIM + t], v[j], w);
        float dv = block_reduce_sum(v1f * v[t], red, t);      // v1f . v
        w -= s1sq * dv * v1f;
        float nr = fmaxf(sqrtf(block_reduce_sum(w * w, red, t)), EPSN);
        v[t] = w / nr;
        __syncthreads();
    }
    vecs[((size_t)b * 4 + 2) * DDIM + t] = v[t];              // v2p

    float a2 = 0.f;
    for (int j = 0; j < DDIM; ++j) a2 = fmaf(Gb[(size_t)j * DDIM + t], v[j], a2);
    float c12 = block_reduce_sum(v1f * v[t], red, t);         // v1f . v2p
    a2 -= s1sq * c12 * v1f;                                   // a2 = G1 v2p
    float nu2sq = block_reduce_sum(v[t] * a2, red, t);
    float nu2   = fmaxf(sqrtf(fmaxf(nu2sq, 0.f)), EPSN);
    float w2    = a2 / nu2;
    float wn2   = sqrtf(block_reduce_sum(w2 * w2, red, t));
    float s2    = wn2 * wn2 / fmaxf(wn2, EPSN);
    float v2f_  = w2 / fmaxf(wn2, EPSN);
    vecs[((size_t)b * 4 + 3) * DDIM + t] = v2f_;

    if (t == 0) {
        float* sb = scal + (size_t)b * 8;
        sb[0] = s1; sb[1] = nu1; sb[2] = s2; sb[3] = nu2; sb[4] = c12;
    }
}

// ---------------------------------------------------------------------------
// Phase 3: fused reconstruction. One thread per column n (u values are
// column-local, no reductions needed).
//   u1[n] = (X v1p)[n] / nu1
//   u2[n] = ((X v2p)[n] - s1*c12*u1[n]) / nu2
//   y[d][n] = sqrt(s1) u1 v1f[d] + sqrt(s2) u2 v2f[d]
//           + (x[d][n] - s1 u1 v1f[d] - s2 u2 v2f[d]) / (sqrt(s2) + 1e-5)
// Output layout (b,d,n) equals the input layout -> coalesced stores.
// ---------------------------------------------------------------------------
__global__ __launch_bounds__(256) void svpn_output(const float* __restrict__ x,
                                                   const float* __restrict__ vecs,
                                                   const float* __restrict__ scal,
                                                   float* __restrict__ out)
{
    __shared__ float Lv1p[DDIM], Lv1f[DDIM], Lv2p[DDIM], Lv2f[DDIM];
    const int b = blockIdx.y;
    const int t = threadIdx.x;
    const int n = blockIdx.x * 256 + t;

    const float* vb = vecs + (size_t)b * 4 * DDIM;
    Lv1p[t] = vb[t];
    Lv1f[t] = vb[DDIM + t];
    Lv2p[t] = vb[2 * DDIM + t];
    Lv2f[t] = vb[3 * DDIM + t];
    __syncthreads();

    const float* sb  = scal + (size_t)b * 8;
    const float  s1  = sb[0], nu1 = sb[1], s2 = sb[2], nu2 = sb[3], c12 = sb[4];
    const float* xb  = x   + (size_t)b * DDIM * (size_t)NDIM;
    float*       ob  = out + (size_t)b * DDIM * (size_t)NDIM;

    float a1 = 0.f, a2 = 0.f;
    #pragma unroll 4
    for (int d = 0; d < DDIM; ++d) {
        float xv = xb[(size_t)d * NDIM + n];
        a1 = fmaf(xv, Lv1p[d], a1);
        a2 = fmaf(xv, Lv2p[d], a2);
    }
    const float u1 = a1 / nu1;
    const float u2 = (a2 - s1 * c12 * u1) / nu2;
    const float k1 = sqrtf(s1) * u1;          // s1^alpha * u1, alpha = 0.5
    const float k2 = sqrtf(s2) * u2;
    const float r  = 1.0f / (sqrtf(s2) + 1e-5f);   // 1 / (s2^(1-alpha) + eps)
    const float su1 = s1 * u1, su2 = s2 * u2;

    #pragma unroll 4
    for (int d = 0; d < DDIM; ++d) {
        float xv  = xb[(size_t)d * NDIM + n];
        float res = xv - su1 * Lv1f[d] - su2 * Lv2f[d];
        ob[(size_t)d * NDIM + n] = k1 * Lv1f[d] + k2 * Lv2f[d] + res * r;
    }
}

// ---------------------------------------------------------------------------
extern "C" void kernel_launch(void* const* d_in, const int* in_sizes, int n_in,
                              void* d_out, int out_size, void* d_ws, size_t ws_size,
                              hipStream_t stream)
{
    (void)in_sizes; (void)n_in; (void)out_size; (void)ws_size;
    const float* x      = (const float*)d_in[0];
    const float* weight = (const float*)d_in[1];
    float*       out    = (float*)d_out;

    // Workspace layout: G (B*D*D = 64 MB) | vecs (B*4*D = 1 MB) | scal (B*8)
    float* G    = (float*)d_ws;
    float* vecs = G + (size_t)BATCH * DDIM * DDIM;
    float* scal = vecs + (size_t)BATCH * 4 * DDIM;

    svpn_syrk_wmma<<<dim3(BATCH), dim3(256), 0, stream>>>(x, G);
    svpn_power   <<<dim3(BATCH), dim3(256), 0, stream>>>(G, weight, vecs, scal);
    svpn_output  <<<dim3(NDIM / 256, BATCH), dim3(256), 0, stream>>>(x, vecs, scal, out);
}